// Net_67542655697471
// MI455X (gfx1250) — compile-verified
//
#include <hip/hip_runtime.h>
#include <hip/hip_bf16.h>
#include <math.h>

typedef float v2f __attribute__((ext_vector_type(2)));
typedef float v4f __attribute__((ext_vector_type(4)));
typedef float v8f __attribute__((ext_vector_type(8)));

#define F_IN 128
#define HID 16

// ---------------------------------------------------------------------------
// degree accumulation: deg[row[e]] += 1
// ---------------------------------------------------------------------------
__global__ void deg_kernel(const long long* __restrict__ row, float* __restrict__ deg,
                           long long nE) {
    long long e = (long long)blockIdx.x * blockDim.x + threadIdx.x;
    if (e < nE) atomicAdd(&deg[row[e]], 1.0f);
}

// deg -> dinv in place:  dinv = deg>0 ? rsqrt(max(deg,1)) : 0
__global__ void dinv_kernel(float* __restrict__ deg, int nN) {
    int i = blockIdx.x * blockDim.x + threadIdx.x;
    if (i < nN) {
        float d = deg[i];
        deg[i] = (d > 0.0f) ? rsqrtf(fmaxf(d, 1.0f)) : 0.0f;
    }
}

// ---------------------------------------------------------------------------
// GEMM1:  xw[N x 32] = x[N x 128] @ [W1[0] | W1[1]]   (fp32 WMMA 16x16x4)
// Block = 128 threads = 4 waves; wave w handles row-tile (blockIdx*4 + w).
// LDS holds Wcat TRANSPOSED (padded stride 132) so each lane's B fragment
// (two consecutive K values) is one b64 LDS load.
// ---------------------------------------------------------------------------
__global__ void gemm1_wmma(const float* __restrict__ x, const float* __restrict__ W1,
                           float* __restrict__ xw, int nN) {
    __shared__ float sBT[32][132];   // ~16.5 KB, padded (even) row stride
    int tid = threadIdx.x;
    for (int i = tid; i < 32 * F_IN; i += 128) {
        int j = i >> 7, f = i & 127;
        // W1 layout (2,128,16): W1[k][f][h] at k*2048 + f*16 + h
        sBT[j][f] = W1[((j >> 4) << 11) + f * 16 + (j & 15)];
    }
    __syncthreads();

    int wave = tid >> 5, lane = tid & 31;
    int m0 = (blockIdx.x * 4 + wave) * 16;
    if (m0 >= nN) return;                      // whole wave exits together
    int M = lane & 15, half = lane >> 4;
    int rowi = m0 + M;
    int rowc = rowi < nN ? rowi : nN - 1;      // clamp; results masked at store
    const float* xrow = x + (size_t)rowc * F_IN;

    v8f acc0 = {}, acc1 = {};
#pragma unroll 8
    for (int k = 0; k < F_IN; k += 4) {
        int c0 = k + 2 * half;
        v2f a  = *(const v2f*)(xrow + c0);          // global_load_b64
        v2f b0 = *(const v2f*)(&sBT[M][c0]);        // ds_load_b64
        v2f b1 = *(const v2f*)(&sBT[16 + M][c0]);   // ds_load_b64
        acc0 = __builtin_amdgcn_wmma_f32_16x16x4_f32(false, a, false, b0, (short)0, acc0, false, false);
        acc1 = __builtin_amdgcn_wmma_f32_16x16x4_f32(false, a, false, b1, (short)0, acc1, false, false);
    }
    if (m0 + 16 <= nN) {            // wave-uniform fast path: no per-lane guards
#pragma unroll
        for (int r = 0; r < 8; ++r) {
            size_t m = (size_t)(m0 + r + 8 * half);
            xw[m * 32 + M]      = acc0[r];
            xw[m * 32 + 16 + M] = acc1[r];
        }
    } else {                        // ragged last tile
#pragma unroll
        for (int r = 0; r < 8; ++r) {
            int m = m0 + r + 8 * half;
            if (m < nN) {
                xw[(size_t)m * 32 + M]      = acc0[r];
                xw[(size_t)m * 32 + 16 + M] = acc1[r];
            }
        }
    }
}

// ---------------------------------------------------------------------------
// Edge scatter: agg[row][j] += (-dinv[row]*dinv[col]) * src[col*stride + off + j]
// one thread per (edge, feature j<16)
// ---------------------------------------------------------------------------
__global__ void scatter16(const float* __restrict__ src, int stride, int off,
                          const long long* __restrict__ row, const long long* __restrict__ col,
                          const float* __restrict__ dinv, float* __restrict__ agg,
                          long long nE) {
    long long t = (long long)blockIdx.x * blockDim.x + threadIdx.x;
    if (t >= nE * 16) return;
    long long e = t >> 4;
    int j = (int)(t & 15);
    long long r = row[e], c = col[e];
    float w = -dinv[r] * dinv[c];
    atomicAdd(&agg[r * 16 + j], w * src[c * (long long)stride + off + j]);
}

// h = relu(xw[:,0:16] + agg + b1)   (vectorized float4)
__global__ void relu_combine(const float* __restrict__ xw, const float* __restrict__ agg,
                             const float* __restrict__ b1, float* __restrict__ h, int nN) {
    int t = blockIdx.x * blockDim.x + threadIdx.x;      // nN*4 threads
    if (t < nN * 4) {
        int i = t >> 2, j4 = (t & 3) * 4;
        v4f xv = *(const v4f*)(xw + (size_t)i * 32 + j4);
        v4f av = *(const v4f*)(agg + (size_t)i * 16 + j4);
        v4f bv = *(const v4f*)(b1 + j4);
        v4f v  = xv + av + bv;
#pragma unroll
        for (int l = 0; l < 4; ++l) v[l] = v[l] > 0.0f ? v[l] : 0.0f;
        *(v4f*)(h + (size_t)i * 16 + j4) = v;
    }
}

// ---------------------------------------------------------------------------
// row-wise log_softmax epilogue for one staged 16x128 tile.
// lane (M, half) handles row M, columns [half*64, half*64+64).
// Partner lanes L and L^16 share the same row -> shuffles always paired.
// ---------------------------------------------------------------------------
__device__ __forceinline__ void softmax_row(const float* __restrict__ rowv,
                                            float* __restrict__ orow, int half) {
    int cbase = half * 64;
    float mx = -INFINITY;
#pragma unroll 8
    for (int c = 0; c < 64; ++c) mx = fmaxf(mx, rowv[cbase + c]);
    mx = fmaxf(mx, __shfl_xor(mx, 16, 32));
    float s = 0.0f;
#pragma unroll 8
    for (int c = 0; c < 64; ++c) s += expf(rowv[cbase + c] - mx);
    s += __shfl_xor(s, 16, 32);
    float lse = mx + logf(s);
#pragma unroll
    for (int c = 0; c < 64; c += 4) {               // global_store_b128
        v4f v = *(const v4f*)(rowv + cbase + c);
        v4f o; o.x = v.x - lse; o.y = v.y - lse; o.z = v.z - lse; o.w = v.w - lse;
        *(v4f*)(orow + cbase + c) = o;
    }
}

// ---------------------------------------------------------------------------
// GEMM2 + fused log_softmax:
//   o[N x 128] = [h | agg2] (N x 32) @ [W2[0]; W2[1]] (32 x 128) + b2
//   out = o - logsumexp(o, axis=1)
// LDS weight tile TRANSPOSED (padded stride 34): B fragment = one b64 load.
// K steps by 4, halves offset by 2 -> a fragment never straddles the h/agg2
// boundary at column 16, and (c0 < 16) <=> (k < 16): compile-time select.
// ---------------------------------------------------------------------------
__global__ void gemm2_softmax(const float* __restrict__ h, const float* __restrict__ agg2,
                              const float* __restrict__ W2, const float* __restrict__ b2,
                              float* __restrict__ out, int nN) {
    __shared__ float sBT[F_IN][34];         // ~17 KB, padded (even) row stride
    __shared__ float sBias[F_IN];
    __shared__ float sOut[4][16][F_IN];     // 32 KB
    int tid = threadIdx.x;
    for (int i = tid; i < 32 * F_IN; i += 128) {
        int r = i >> 7, c = i & 127;
        // W2 layout (2,16,128): W2[k][hh][f] at k*2048 + hh*128 + f
        sBT[c][r] = W2[((r >> 4) << 11) + (r & 15) * F_IN + c];
    }
    if (tid < F_IN) sBias[tid] = b2[tid];
    __syncthreads();

    int wave = tid >> 5, lane = tid & 31;
    int m0 = (blockIdx.x * 4 + wave) * 16;
    bool active = m0 < nN;
    bool full   = m0 + 16 <= nN;            // wave-uniform
    int M = lane & 15, half = lane >> 4;
    int rowi = m0 + M;

    if (active) {
        int rowc = rowi < nN ? rowi : nN - 1;   // clamp; masked at final store
        long long rb = (long long)rowc * HID;
        v8f acc[8];
#pragma unroll
        for (int t = 0; t < 8; ++t) acc[t] = (v8f){};
#pragma unroll
        for (int k = 0; k < 32; k += 4) {
            int c0 = k + 2 * half;
            // compile-time source select: k<16 -> h, else agg2 (c0-16)
            const float* asrc = (k < 16) ? (h + rb + c0) : (agg2 + rb + (c0 - 16));
            v2f a = *(const v2f*)asrc;              // global_load_b64
#pragma unroll
            for (int t = 0; t < 8; ++t) {
                v2f b = *(const v2f*)(&sBT[t * 16 + M][c0]);   // ds_load_b64
                acc[t] = __builtin_amdgcn_wmma_f32_16x16x4_f32(false, a, false, b, (short)0, acc[t], false, false);
            }
        }
        // stage this wave's 16x128 tile (+bias) into its own LDS region
#pragma unroll
        for (int t = 0; t < 8; ++t)
#pragma unroll
            for (int r = 0; r < 8; ++r)
                sOut[wave][r + 8 * half][t * 16 + M] = acc[t][r] + sBias[t * 16 + M];
    }
    __syncthreads();   // all 128 threads reach this; makes staged tiles visible

    if (active) {
        const float* rowv = &sOut[wave][M][0];
        float* orow = out + (size_t)rowi * F_IN;
        if (full) {
            softmax_row(rowv, orow, half);          // uniform, unguarded path
        } else if (rowi < nN) {
            softmax_row(rowv, orow, half);          // ragged last tile
        }
    }
}

// ---------------------------------------------------------------------------
extern "C" void kernel_launch(void* const* d_in, const int* in_sizes, int n_in,
                              void* d_out, int out_size, void* d_ws, size_t ws_size,
                              hipStream_t stream) {
    const float* x      = (const float*)d_in[0];
    const long long* ei = (const long long*)d_in[1];   // int64 (2, E)
    const float* W1     = (const float*)d_in[2];
    const float* b1     = (const float*)d_in[3];
    const float* W2     = (const float*)d_in[4];
    const float* b2     = (const float*)d_in[5];
    float* out          = (float*)d_out;

    int       nN = in_sizes[0] / F_IN;
    long long nE = (long long)in_sizes[1] / 2;
    const long long* row = ei;
    const long long* col = ei + nE;

    // workspace layout (floats), offsets padded to 64-float multiples:
    //   dinv | xw(32S) | agg(16S) | h(16S)   (S = padded N)
    size_t S = ((size_t)nN + 63) & ~(size_t)63;
    float* ws   = (float*)d_ws;
    float* dinv = ws;
    float* xw   = ws + S;
    float* agg  = ws + 33 * S;
    float* hbuf = ws + 49 * S;

    hipMemsetAsync(dinv, 0, (size_t)nN * sizeof(float), stream);
    hipMemsetAsync(agg,  0, (size_t)16 * nN * sizeof(float), stream);

    {   // degrees + normalization
        int blk = 256;
        long long g = (nE + blk - 1) / blk;
        deg_kernel<<<(unsigned)g, blk, 0, stream>>>(row, dinv, nE);
        dinv_kernel<<<(nN + 255) / 256, 256, 0, stream>>>(dinv, nN);
    }

    int tiles  = (nN + 15) / 16;
    int blocks = (tiles + 3) / 4;
    gemm1_wmma<<<blocks, 128, 0, stream>>>(x, W1, xw, nN);

    {   // layer-1 scatter on 16 projected features (xw cols 16..31)
        long long work = nE * 16;
        long long g = (work + 255) / 256;
        scatter16<<<(unsigned)g, 256, 0, stream>>>(xw, 32, 16, row, col, dinv, agg, nE);
    }

    relu_combine<<<(nN * 4 + 255) / 256, 256, 0, stream>>>(xw, agg, b1, hbuf, nN);

    hipMemsetAsync(agg, 0, (size_t)16 * nN * sizeof(float), stream);

    {   // layer-2 scatter on the 16 hidden features
        long long work = nE * 16;
        long long g = (work + 255) / 256;
        scatter16<<<(unsigned)g, 256, 0, stream>>>(hbuf, 16, 0, row, col, dinv, agg, nE);
    }

    gemm2_softmax<<<blocks, 128, 0, stream>>>(hbuf, agg, W2, b2, out, nN);
}